// BoxFilter_5669356832880
// MI455X (gfx1250) — compile-verified
//
#include <hip/hip_runtime.h>

// Fused separable box filter (r = 8, k = 17) for [N=8, C=32, H=512, W=512] f32.
// Both 1D passes are banded matmuls executed with V_WMMA_F32_16X16X4_F32
// (full f32 precision, matching the f32 reference).
//
//   horizontal:  Y(16x16) = X(16x32)     * Bband(32x16),  Bband[K][n] = 1/17 iff K-n in [0,16]
//   vertical:    Z(16x16) = Aband(16x32) * H(32x16),      Aband[m][K] = 1/17 iff K-m in [0,16]
//
// The 17-tap window for 16 outputs spans exactly K = 32 -> 8 chained
// 16x16x4 f32 WMMAs per sub-tile. Band operands are generated in registers
// from the ISA-documented f32 A/B lane layouts (lane L: idx = L&15,
// K = 2*(L>>4) + vgpr); they never touch memory.

typedef float v2f __attribute__((ext_vector_type(2)));
typedef float v4f __attribute__((ext_vector_type(4)));
typedef float v8f __attribute__((ext_vector_type(8)));

#define RAD      8
#define IMG      512
#define TILE     64
#define HALO     80          // TILE + 2*RAD
#define IN_PITCH 84          // mult of 4 -> 16B-aligned b128 LDS stores; 84%64=20 has
                             // mod-64 cycle length 16 -> conflict-free 16-row column reads
#define H_PITCH  66          // 8-row half-wave offset maps to +16 banks -> conflict-free D writes

__device__ __forceinline__ float band_val(int K, int idx, float inv) {
    // 1/17 if (K - idx) in [0, 2*RAD], else 0
    return ((unsigned)(K - idx) <= 2u * RAD) ? inv : 0.0f;
}

__global__ __launch_bounds__(256) void box_filter_wmma(
    const float* __restrict__ in, float* __restrict__ out)
{
    __shared__ float lds_in[HALO * IN_PITCH]; // 80x80 halo tile (zero padded)
    __shared__ float lds_h[HALO * H_PITCH];   // horizontally filtered, vertical halo kept

    const float inv = 1.0f / 17.0f;
    const int tid  = threadIdx.x;
    const int lane = tid & 31;
    // Force wave index into an SGPR so sub-tile loops compile as scalar
    // branches: EXEC stays all-ones through every WMMA (ISA requirement).
    const int wave = __builtin_amdgcn_readfirstlane(tid >> 5);
    const int half = lane >> 4;      // which 16-lane half of the wave
    const int l16  = lane & 15;      // M (A-operand) or N (B/C/D-operand) index
    const int kk0  = half << 1;      // K offset of VGPR0 within a K=4 chunk

    const int x0  = blockIdx.x * TILE;
    const int y0  = blockIdx.y * TILE;
    const int img = blockIdx.z;      // fused N*C index
    const float* __restrict__ src = in  + (size_t)img * IMG * IMG;
    float*       __restrict__ dst = out + (size_t)img * IMG * IMG;

    // ---- Phase 1: global -> LDS, 80x80 halo as 80x20 aligned float4 groups.
    // Halo row start x0-8 is a multiple of 4 floats -> 16B-aligned b128 loads;
    // element-wise fallback only on image-boundary groups.
    for (int t = tid; t < HALO * (HALO / 4); t += 256) {
        int ly  = t / (HALO / 4);
        int g   = t - ly * (HALO / 4);
        int gy  = y0 - RAD + ly;
        int gxb = x0 - RAD + 4 * g;
        v4f v = {0.0f, 0.0f, 0.0f, 0.0f};
        if (gy >= 0 && gy < IMG) {
            const float* rowp = src + (size_t)gy * IMG;
            if (gxb >= 0 && gxb + 3 < IMG) {
                v = *(const v4f*)(rowp + gxb);          // global_load_b128
            } else {
                #pragma unroll
                for (int j = 0; j < 4; ++j) {
                    int gx = gxb + j;
                    if (gx >= 0 && gx < IMG) v[j] = rowp[gx];
                }
            }
        }
        *(v4f*)(&lds_in[ly * IN_PITCH + 4 * g]) = v;    // ds_store_b128 (16B aligned)
    }
    __syncthreads();

    // ---- Phase 2: horizontal pass. 20 sub-tiles (5 row-blocks x 4 col-blocks), 8 WMMAs each.
    for (int st = wave; st < 20; st += 8) {
        int rb = st >> 2;
        int cb = st & 3;
        v8f acc = {};
        int arow = (rb * 16 + l16) * IN_PITCH + cb * 16;   // A: X rows from lds_in
        #pragma unroll
        for (int kc = 0; kc < 8; ++kc) {
            int K0 = kc * 4 + kk0;
            v2f a, b;
            a.x = lds_in[arow + K0];
            a.y = lds_in[arow + K0 + 1];
            b.x = band_val(K0,     l16, inv);              // B: band matrix, in-register
            b.y = band_val(K0 + 1, l16, inv);
            acc = __builtin_amdgcn_wmma_f32_16x16x4_f32(
                false, a, false, b, (short)0, acc, false, false);
        }
        // D layout: VGPR v -> row v (lanes 0-15) / row v+8 (lanes 16-31), col = l16
        int hbase = (rb * 16 + 8 * half) * H_PITCH + cb * 16 + l16;
        #pragma unroll
        for (int v = 0; v < 8; ++v)
            lds_h[hbase + v * H_PITCH] = acc[v];
    }
    __syncthreads();

    // ---- Phase 3: vertical pass. 16 sub-tiles (4x4), 8 WMMAs each, stores to global.
    for (int st = wave; st < 16; st += 8) {
        int vb = st >> 2;
        int cb = st & 3;
        v8f acc = {};
        int hcol = cb * 16 + l16;
        #pragma unroll
        for (int kc = 0; kc < 8; ++kc) {
            int K0 = kc * 4 + kk0;
            v2f a, b;
            a.x = band_val(K0,     l16, inv);              // A: band matrix, in-register
            a.y = band_val(K0 + 1, l16, inv);
            b.x = lds_h[(vb * 16 + K0)     * H_PITCH + hcol];
            b.y = lds_h[(vb * 16 + K0 + 1) * H_PITCH + hcol];
            acc = __builtin_amdgcn_wmma_f32_16x16x4_f32(
                false, a, false, b, (short)0, acc, false, false);
        }
        int orow = y0 + vb * 16 + 8 * half;
        int ocol = x0 + cb * 16 + l16;
        #pragma unroll
        for (int v = 0; v < 8; ++v)
            dst[(size_t)(orow + v) * IMG + ocol] = acc[v]; // 16-lane contiguous 64B stores
    }
}

extern "C" void kernel_launch(void* const* d_in, const int* in_sizes, int n_in,
                              void* d_out, int out_size, void* d_ws, size_t ws_size,
                              hipStream_t stream) {
    (void)n_in; (void)out_size; (void)d_ws; (void)ws_size;
    const float* x = (const float*)d_in[0];
    float* y = (float*)d_out;
    const int images = in_sizes[0] / (IMG * IMG);          // N*C = 256
    dim3 grid(IMG / TILE, IMG / TILE, images);             // (8, 8, 256)
    box_filter_wmma<<<grid, 256, 0, stream>>>(x, y);
}